// MeanAggregator_f_2551210574181
// MI455X (gfx1250) — compile-verified
//
#include <hip/hip_runtime.h>

// MI455X / gfx1250, wave32. See analysis: phase 2 (edge scatter into L2-resident
// buffers) dominates; phase 1 uses V_WMMA_F32_16X16X4_F32 for the fused MLP.

typedef __attribute__((ext_vector_type(2))) float v2f;
typedef __attribute__((ext_vector_type(8))) float v8f;

#define NNODES 50000
#define FEAT   128

// ---------------------------------------------------------------------------
// Kernel 1: H = tanh(X @ W1 + b1) @ W2 + b2   (N x 128, all fp32, WMMA)
// 256 threads = 8 waves. Block owns 16 rows; wave w owns columns [16w,16w+16).
// ---------------------------------------------------------------------------
__global__ __launch_bounds__(256) void gnn_mlp_wmma(
    const float* __restrict__ X,  const float* __restrict__ W1,
    const float* __restrict__ b1, const float* __restrict__ W2,
    const float* __restrict__ b2, float* __restrict__ H)
{
  // pad rows to 132 floats: stride 132 % 64 banks = 4 -> conflict-free column reads
  __shared__ alignas(16) float xs[16][132];
  __shared__ alignas(16) float ts[16][132];

  const int tid   = threadIdx.x;
  const int wave  = tid >> 5;
  const int lane  = tid & 31;
  const int laneM = lane & 15;          // M (A-frag) or N (B/C/D-frag)
  const int hiK   = (lane >> 4) << 1;   // lanes 16-31 hold K+2 per ISA A/B layout
  const int mbase = (lane >> 4) << 3;   // C/D: lanes 16-31 hold rows M+8
  const int col0  = wave * 16;
  const int row0  = blockIdx.x * 16;    // N = 50000 = 3125 * 16, no remainder

  // Stage 16x128 X tile into LDS, float4-coalesced (512 float4 / 256 threads)
  {
    const float4* src4 = (const float4*)(X + (size_t)row0 * FEAT);
    #pragma unroll
    for (int j = tid; j < 16 * 32; j += 256) {
      const int r = j >> 5, c4 = j & 31;
      *(float4*)&xs[r][c4 * 4] = src4[j];
    }
  }
  __syncthreads();

  // ---- GEMM1: acc = X_tile @ W1[:, col0:col0+16], K = 128 in steps of 4 ----
  v8f acc = {};
  #pragma unroll 4
  for (int kk = 0; kk < 32; ++kk) {
    const int k0 = kk * 4 + hiK;
    v2f a, b;
    a.x = xs[laneM][k0];
    a.y = xs[laneM][k0 + 1];
    b.x = W1[(size_t)k0 * FEAT + col0 + laneM];
    b.y = W1[(size_t)(k0 + 1) * FEAT + col0 + laneM];
    acc = __builtin_amdgcn_wmma_f32_16x16x4_f32(false, a, false, b,
                                                (short)0, acc, false, false);
  }
  // bias + tanh, write t tile to LDS per C/D layout (VGPR r -> rows r / r+8)
  {
    const float bias = b1[col0 + laneM];
    #pragma unroll
    for (int r = 0; r < 8; ++r)
      ts[mbase + r][col0 + laneM] = tanhf(acc[r] + bias);
  }
  __syncthreads();

  // ---- GEMM2: acc = t_tile @ W2[:, col0:col0+16] ----
  acc = (v8f){};
  #pragma unroll 4
  for (int kk = 0; kk < 32; ++kk) {
    const int k0 = kk * 4 + hiK;
    v2f a, b;
    a.x = ts[laneM][k0];
    a.y = ts[laneM][k0 + 1];
    b.x = W2[(size_t)k0 * FEAT + col0 + laneM];
    b.y = W2[(size_t)(k0 + 1) * FEAT + col0 + laneM];
    acc = __builtin_amdgcn_wmma_f32_16x16x4_f32(false, a, false, b,
                                                (short)0, acc, false, false);
  }
  {
    const float bias = b2[col0 + laneM];
    #pragma unroll
    for (int r = 0; r < 8; ++r)
      H[(size_t)(row0 + mbase + r) * FEAT + col0 + laneM] = acc[r] + bias;
  }
}

// ---------------------------------------------------------------------------
// Kernel 2: one wave per edge. value = (src==dst) ? mask[ind] : 1.
// Skip zero-valued edges (adds of 0 are no-ops for both segment sums).
// Each lane: float4 of h[dst] -> 4x global_atomic_add_f32 into results[src].
// ---------------------------------------------------------------------------
__global__ __launch_bounds__(256) void gnn_edge_agg(
    const int* __restrict__ edges, const float* __restrict__ H,
    float* __restrict__ results, float* __restrict__ row_sum,
    const int* __restrict__ ind_p, int E)
{
  const int gtid = blockIdx.x * blockDim.x + threadIdx.x;
  const int wid  = gtid >> 5;
  const int lane = gtid & 31;
  if (wid >= E) return;

  const int src = edges[2 * wid];        // uniform across the wave
  const int dst = edges[2 * wid + 1];
  const int ind = *ind_p;                // mask = {1,1,0,0}
  const float mask_v = (ind < 2) ? 1.0f : 0.0f;
  const float v = (src == dst) ? mask_v : 1.0f;
  if (v != 0.0f) {                       // wave-uniform branch
    if (lane == 0) atomicAdd(&row_sum[src], v);
    const float4 hv = *(const float4*)(H + (size_t)dst * FEAT + lane * 4);
    float* rp = results + (size_t)src * FEAT + lane * 4;
    atomicAdd(rp + 0, v * hv.x);
    atomicAdd(rp + 1, v * hv.y);
    atomicAdd(rp + 2, v * hv.z);
    atomicAdd(rp + 3, v * hv.w);
  }
}

// ---------------------------------------------------------------------------
// Kernel 3: out[b, f] = results[node_idx[b], f] / max-fix(row_sum)
// ---------------------------------------------------------------------------
__global__ __launch_bounds__(256) void gnn_gather(
    const int* __restrict__ node_idx, const float* __restrict__ results,
    const float* __restrict__ row_sum, float* __restrict__ out, int Bn)
{
  const int idx = blockIdx.x * blockDim.x + threadIdx.x;
  if (idx >= Bn * FEAT) return;
  const int b = idx >> 7, f = idx & (FEAT - 1);
  const int n = node_idx[b];
  float rs = row_sum[n];
  rs = (rs == 0.0f) ? 1.0f : rs;
  out[idx] = results[(size_t)n * FEAT + f] / rs;
}

extern "C" void kernel_launch(void* const* d_in, const int* in_sizes, int n_in,
                              void* d_out, int out_size, void* d_ws, size_t ws_size,
                              hipStream_t stream) {
  const int*   edges    = (const int*)d_in[0];
  const int*   node_idx = (const int*)d_in[1];
  const float* X        = (const float*)d_in[2];
  const float* W1       = (const float*)d_in[3];
  const float* b1       = (const float*)d_in[4];
  const float* W2       = (const float*)d_in[5];
  const float* b2       = (const float*)d_in[6];
  const int*   ind      = (const int*)d_in[7];

  const int E  = in_sizes[0] / 2;
  const int Bn = in_sizes[1];

  // Workspace layout: H | results | row_sum  (contiguous, ~51.4 MB)
  float* H       = (float*)d_ws;
  float* results = H + (size_t)NNODES * FEAT;
  float* row_sum = results + (size_t)NNODES * FEAT;

  // zero the accumulators every call (results + row_sum are contiguous)
  hipMemsetAsync(results, 0,
                 ((size_t)NNODES * FEAT + NNODES) * sizeof(float), stream);

  gnn_mlp_wmma<<<NNODES / 16, 256, 0, stream>>>(X, W1, b1, W2, b2, H);
  gnn_edge_agg<<<(E * 32 + 255) / 256, 256, 0, stream>>>(edges, H, results,
                                                         row_sum, ind, E);
  gnn_gather<<<(Bn * FEAT + 255) / 256, 256, 0, stream>>>(node_idx, results,
                                                          row_sum, (float*)d_out, Bn);
}